// GCN_3951369912440
// MI455X (gfx1250) — compile-verified
//
#include <hip/hip_runtime.h>
#include <math.h>

// ---------------------------------------------------------------------------
// GCN forward (2-layer) for MI455X / gfx1250.
//
// Memory-bound over the 16M-edge list; all per-node tensors (<=32MB each) are
// L2-resident (192MB L2), so the gather + scatter-atomic traffic stays
// on-chip and only the 3 streaming passes over the 128MB edge index hit HBM
// (~16us at 23.3 TB/s floor). Dense GEMMs use V_WMMA_F32_16X16X4_F32
// (exact f32; wave32 operand layout per CDNA5 ISA §7.12.2),
// with branchless zero-padding and wave-uniform tail handling so the WMMA
// path is straight-line code.
// ---------------------------------------------------------------------------

typedef __attribute__((ext_vector_type(2))) float v2f;
typedef __attribute__((ext_vector_type(8))) float v8f;

__device__ __forceinline__ void fatomic_add(float* p, float v) {
    // lowers to global_atomic_add_f32 (no CAS loop)
    unsafeAtomicAdd(p, v);
}

// ---- init: deg=1 (self loop), agg1=0, agg2=0 ------------------------------
__global__ void k_init(float* __restrict__ deg, float* __restrict__ agg1,
                       float* __restrict__ agg2, int N) {
    int i = blockIdx.x * blockDim.x + threadIdx.x;
    if (i < N * 16) agg1[i] = 0.0f;
    if (i < N * 8)  agg2[i] = 0.0f;
    if (i < N)      deg[i]  = 1.0f;
}

// ---- in-degree over edge targets ------------------------------------------
__global__ void k_degree(const int* __restrict__ col, float* __restrict__ deg, int E) {
    int e = blockIdx.x * blockDim.x + threadIdx.x;
    if (e < E) fatomic_add(&deg[col[e]], 1.0f);
}

__global__ void k_rsqrt(float* __restrict__ deg, int N) {
    int i = blockIdx.x * blockDim.x + threadIdx.x;
    if (i < N) deg[i] = rsqrtf(deg[i]);   // deg >= 1 always (self loops)
}

// ---- layer-1 GEMM: h1[N,16] = x[N,3] @ W1[3,16] via WMMA f32 16x16x4 -------
// A layout (32-bit A 16x4): lanes 0-15 hold {K0,K1} of row M=lane,
// lanes 16-31 hold {K2,K3} of row M=lane-16. K=3 column is a zero pad,
// realized branchlessly (in-bounds load * 0/1 select, no EXEC games).
__global__ void k_gemm1(const float* __restrict__ x, const float* __restrict__ W1,
                        float* __restrict__ h1, int N) {
    int wave = threadIdx.x >> 5;
    int lane = threadIdx.x & 31;
    int tile = blockIdx.x * (blockDim.x >> 5) + wave;
    int numTiles = (N + 15) >> 4;
    if (tile >= numTiles) return;          // wave-uniform exit, EXEC stays all-1s
    int m0   = tile << 4;
    int half = lane >> 4;                  // 0: K={0,1}  1: K={2,3}
    int n    = lane & 15;
    int m    = m0 + n; if (m >= N) m = N - 1;   // clamp loads on tail tile
    float pad = half ? 0.0f : 1.0f;

    v2f a;
    a.x = x[(size_t)m * 3 + 2 * half];               // K0 / K2
    a.y = x[(size_t)m * 3 + 1 + half] * pad;         // K1 / zero-pad K3
    v2f b;
    b.x = W1[(half ? 2 : 0) * 16 + n];               // W1 row K0 / K2
    b.y = W1[(half ? 2 : 1) * 16 + n] * pad;         // W1 row K1 / zero-pad K3

    v8f c = {};
    c = __builtin_amdgcn_wmma_f32_16x16x4_f32(false, a, false, b, (short)0, c, false, false);

    // D: VGPR j -> row m0+j (lanes 0-15) / m0+j+8 (lanes 16-31), col n.
    float* dst = h1 + (size_t)(m0 + half * 8) * 16 + n;
    if (m0 + 16 <= N) {                    // wave-uniform fast path
#pragma unroll
        for (int j = 0; j < 8; ++j) dst[(size_t)j * 16] = c[j];
    } else {
#pragma unroll
        for (int j = 0; j < 8; ++j)
            if (m0 + half * 8 + j < N) dst[(size_t)j * 16] = c[j];
    }
}

// ---- layer-1 edge scatter: agg1[t] += dis[s]*dis[t] * h1[s] ---------------
__global__ void k_scatter1(const int* __restrict__ row, const int* __restrict__ col,
                           const float* __restrict__ dis, const float* __restrict__ h1,
                           float* __restrict__ agg1, int E) {
    int e = blockIdx.x * blockDim.x + threadIdx.x;
    if (e >= E) return;
    int s = row[e], t = col[e];
    float w = dis[s] * dis[t];
    const float4* hs = (const float4*)(h1 + (size_t)s * 16);
    float* dst = agg1 + (size_t)t * 16;
#pragma unroll
    for (int q = 0; q < 4; ++q) {
        float4 v = hs[q];
        fatomic_add(dst + q * 4 + 0, w * v.x);
        fatomic_add(dst + q * 4 + 1, w * v.y);
        fatomic_add(dst + q * 4 + 2, w * v.z);
        fatomic_add(dst + q * 4 + 3, w * v.w);
    }
}

// ---- self-loop + bias + ReLU (in place: agg1 -> y1) -----------------------
__global__ void k_post1(const float* __restrict__ h1, const float* __restrict__ dis,
                        const float* __restrict__ b1, float* __restrict__ agg1, int N) {
    int i = blockIdx.x * blockDim.x + threadIdx.x;
    if (i >= N * 16) return;
    int t = i >> 4, f = i & 15;
    float d = dis[t];
    float v = agg1[i] + d * d * h1[i] + b1[f];
    agg1[i] = v > 0.0f ? v : 0.0f;
}

// ---- layer-2 GEMM: h2[N,8] = y1[N,16] @ W2pad[16,16] via 4 chained WMMAs --
// B (16x7) is zero-padded to 16x16 in LDS once per block, so the K-loop
// reads it with unconditional ds_load (no divergent predicated global loads).
// Output cols 7..15 multiply zero B columns, so h2's stride-8 pad col is 0.
__global__ void k_gemm2(const float* __restrict__ y1, const float* __restrict__ W2,
                        float* __restrict__ h2, int N) {
    __shared__ float w2p[256];             // [k][n] 16x16, cols 7..15 = 0
    {
        int rk = threadIdx.x >> 4, cc = threadIdx.x & 15;
        w2p[threadIdx.x] = (cc < 7) ? W2[rk * 7 + cc] : 0.0f;
    }
    __syncthreads();                       // before any wave-uniform exit

    int wave = threadIdx.x >> 5;
    int lane = threadIdx.x & 31;
    int tile = blockIdx.x * (blockDim.x >> 5) + wave;
    int numTiles = (N + 15) >> 4;
    if (tile >= numTiles) return;
    int m0   = tile << 4;
    int half = lane >> 4;
    int n    = lane & 15;
    int m    = m0 + n; if (m >= N) m = N - 1;

    const float4* yr = (const float4*)(y1 + (size_t)m * 16);   // 4x b128 row load
    v8f c = {};
#pragma unroll
    for (int kc = 0; kc < 4; ++kc) {
        float4 r = yr[kc];
        int k0 = kc * 4 + half * 2;        // this half-wave's K pair
        v2f a;
        a.x = half ? r.z : r.x;            // branchless half-wave select
        a.y = half ? r.w : r.y;
        v2f b;
        b.x = w2p[k0 * 16 + n];
        b.y = w2p[(k0 + 1) * 16 + n];
        c = __builtin_amdgcn_wmma_f32_16x16x4_f32(false, a, false, b, (short)0, c, false, false);
    }

    float* dst = h2 + (size_t)(m0 + half * 8) * 8 + n;
    if (m0 + 16 <= N) {                    // wave-uniform fast path
        if (n < 8) {                       // single divergence around 8 stores
#pragma unroll
            for (int j = 0; j < 8; ++j) dst[(size_t)j * 8] = c[j];
        }
    } else {
#pragma unroll
        for (int j = 0; j < 8; ++j)
            if (n < 8 && m0 + half * 8 + j < N) dst[(size_t)j * 8] = c[j];
    }
}

// ---- layer-2 edge scatter: agg2[t] += dis[s]*dis[t] * h2[s] ---------------
__global__ void k_scatter2(const int* __restrict__ row, const int* __restrict__ col,
                           const float* __restrict__ dis, const float* __restrict__ h2,
                           float* __restrict__ agg2, int E) {
    int e = blockIdx.x * blockDim.x + threadIdx.x;
    if (e >= E) return;
    int s = row[e], t = col[e];
    float w = dis[s] * dis[t];
    const float4* hs = (const float4*)(h2 + (size_t)s * 8);
    float4 v0 = hs[0];
    float4 v1 = hs[1];
    float* dst = agg2 + (size_t)t * 8;
    fatomic_add(dst + 0, w * v0.x);
    fatomic_add(dst + 1, w * v0.y);
    fatomic_add(dst + 2, w * v0.z);
    fatomic_add(dst + 3, w * v0.w);
    fatomic_add(dst + 4, w * v1.x);
    fatomic_add(dst + 5, w * v1.y);
    fatomic_add(dst + 6, w * v1.z);
    // feature 7 is the zero pad; skip
}

// ---- self-loop + bias + log_softmax ---------------------------------------
__global__ void k_final(const float* __restrict__ agg2, const float* __restrict__ h2,
                        const float* __restrict__ dis, const float* __restrict__ b2,
                        float* __restrict__ out, int N) {
    int t = blockIdx.x * blockDim.x + threadIdx.x;
    if (t >= N) return;
    float d  = dis[t];
    float d2 = d * d;
    float v[7];
    float mx = -3.402823466e38f;
#pragma unroll
    for (int f = 0; f < 7; ++f) {
        v[f] = agg2[(size_t)t * 8 + f] + d2 * h2[(size_t)t * 8 + f] + b2[f];
        mx = fmaxf(mx, v[f]);
    }
    float sum = 0.0f;
#pragma unroll
    for (int f = 0; f < 7; ++f) sum += __expf(v[f] - mx);
    float lse = __logf(sum);
#pragma unroll
    for (int f = 0; f < 7; ++f) out[(size_t)t * 7 + f] = v[f] - mx - lse;
}

// ---------------------------------------------------------------------------
extern "C" void kernel_launch(void* const* d_in, const int* in_sizes, int n_in,
                              void* d_out, int out_size, void* d_ws, size_t ws_size,
                              hipStream_t stream) {
    const float* x  = (const float*)d_in[0];
    const int*   ei = (const int*)d_in[1];     // [2, E] int32 (harness convention)
    const float* W1 = (const float*)d_in[2];
    const float* b1 = (const float*)d_in[3];
    const float* W2 = (const float*)d_in[4];
    const float* b2 = (const float*)d_in[5];
    float* out = (float*)d_out;

    int N = in_sizes[0] / 3;
    int E = in_sizes[1] / 2;
    const int* row = ei;       // sources
    const int* col = ei + E;   // targets

    // workspace carve-up: deg/dis | h1 | agg1(->y1) | h2 | agg2   (~98MB)
    char* ws = (char*)d_ws;
    size_t off = 0;
    float* deg  = (float*)(ws + off); off += (size_t)N * 4;
    float* h1   = (float*)(ws + off); off += (size_t)N * 16 * 4;
    float* agg1 = (float*)(ws + off); off += (size_t)N * 16 * 4;
    float* h2   = (float*)(ws + off); off += (size_t)N * 8 * 4;
    float* agg2 = (float*)(ws + off); off += (size_t)N * 8 * 4;
    (void)ws_size; (void)n_in; (void)out_size;

    const int B = 256;                       // 8 waves per block (wave32)
    int gInit  = (N * 16 + B - 1) / B;
    int gEdge  = (E + B - 1) / B;
    int gNode  = (N + B - 1) / B;
    int gTiles = (((N + 15) / 16) + (B / 32) - 1) / (B / 32);

    k_init    <<<gInit,  B, 0, stream>>>(deg, agg1, agg2, N);
    k_degree  <<<gEdge,  B, 0, stream>>>(col, deg, E);
    k_rsqrt   <<<gNode,  B, 0, stream>>>(deg, N);                 // deg -> dis
    k_gemm1   <<<gTiles, B, 0, stream>>>(x, W1, h1, N);
    k_scatter1<<<gEdge,  B, 0, stream>>>(row, col, deg, h1, agg1, E);
    k_post1   <<<gInit,  B, 0, stream>>>(h1, deg, b1, agg1, N);   // agg1 -> y1
    k_gemm2   <<<gTiles, B, 0, stream>>>(agg1, W2, h2, N);
    k_scatter2<<<gEdge,  B, 0, stream>>>(row, col, deg, h2, agg2, E);
    k_final   <<<gNode,  B, 0, stream>>>(agg2, h2, deg, b2, out, N);
}